// BagModel_4904852652363
// MI455X (gfx1250) — compile-verified
//
#include <hip/hip_runtime.h>

typedef __attribute__((ext_vector_type(2))) float v2f;
typedef __attribute__((ext_vector_type(8))) float v8f;

namespace {
constexpr int kDIn          = 32;
constexpr int kDHid         = 64;
constexpr int kBagSize      = 20;
constexpr int kBagsPerGroup = 4;
constexpr int kRowsPerGroup = 80;   // lcm(16-row WMMA tile, 20-row bag)
constexpr int kTilesPerGroup= 5;
constexpr int kWavesPerBlk  = 8;
constexpr int kBlockSize    = kWavesPerBlk * 32;
}

__global__ __launch_bounds__(kBlockSize)
void bag_mil_wmma_kernel(const float* __restrict__ x,
                         const int*   __restrict__ ids,
                         const float* __restrict__ W1,
                         const float* __restrict__ b1,
                         const float* __restrict__ W2,
                         const float* __restrict__ b2,
                         float* __restrict__ out,
                         int N, int B, int ngroups)
{
    __shared__ float s_buf[kWavesPerBlk][kRowsPerGroup];

    const int lane = (int)(threadIdx.x & 31u);
    const int wave = (int)(threadIdx.x >> 5);
    const int hl   = lane >> 4;      // 0: lanes 0-15, 1: lanes 16-31
    const int m    = lane & 15;

    // ---- Wave-invariant operands, held in VGPRs for the whole kernel ----
    // B fragments of W1 for V_WMMA_F32_16X16X4_F32.
    // nb = 16-column block of W1 (N dim), kk = K-step of 4.
    // lanes 0-15 carry K = {0,1} in {.x,.y}; lanes 16-31 carry K = {2,3}.
    v2f bfrag[4][8];
    #pragma unroll
    for (int nb = 0; nb < 4; ++nb) {
        const int col = nb * 16 + m;
        #pragma unroll
        for (int kk = 0; kk < 8; ++kk) {
            const int k0 = kk * 4 + hl * 2;
            bfrag[nb][kk].x = W1[(k0 + 0) * kDHid + col];
            bfrag[nb][kk].y = W1[(k0 + 1) * kDHid + col];
        }
    }
    float b1v[4], w2v[4];
    #pragma unroll
    for (int nb = 0; nb < 4; ++nb) {
        b1v[nb] = b1[nb * 16 + m];
        w2v[nb] = W2[nb * 16 + m];
    }
    const float bias2   = b2[0];
    const float inv_cnt = 1.0f / (float)kBagSize;

    const int waveGlobal = (int)((blockIdx.x * blockDim.x + threadIdx.x) >> 5);
    const int nWaves     = (int)((gridDim.x * blockDim.x) >> 5);

    for (int g = waveGlobal; g < ngroups; g += nWaves) {
        const int r0 = g * kRowsPerGroup;

        #pragma unroll
        for (int t = 0; t < kTilesPerGroup; ++t) {
            const int tr0 = r0 + t * 16;

            // C initialized with b1 broadcast down each column block.
            v8f acc[4];
            #pragma unroll
            for (int nb = 0; nb < 4; ++nb)
                #pragma unroll
                for (int v = 0; v < 8; ++v)
                    acc[nb][v] = b1v[nb];

            // A fragment source row for this lane (clamped for the tail).
            int arow = tr0 + m;
            if (arow > N - 1) arow = N - 1;
            const float* xr = x + (size_t)arow * kDIn;

            #pragma unroll
            for (int kk = 0; kk < 8; ++kk) {
                v2f a;
                a.x = xr[kk * 4 + hl * 2 + 0];
                a.y = xr[kk * 4 + hl * 2 + 1];
                #pragma unroll
                for (int nb = 0; nb < 4; ++nb) {
                    acc[nb] = __builtin_amdgcn_wmma_f32_16x16x4_f32(
                        false, a, false, bfrag[nb][kk],
                        (short)0, acc[nb], false, false);
                }
            }

            // relu + per-column W2 weight, partial dot per lane.
            // D layout: VGPR v -> row (v + 8*hl), column nb*16 + m.
            float p[8];
            #pragma unroll
            for (int v = 0; v < 8; ++v) p[v] = 0.0f;
            #pragma unroll
            for (int nb = 0; nb < 4; ++nb) {
                #pragma unroll
                for (int v = 0; v < 8; ++v) {
                    float hv = acc[nb][v];
                    hv = hv > 0.0f ? hv : 0.0f;
                    p[v] += hv * w2v[nb];
                }
            }

            // 16-lane butterfly reduction (stays within each half of wave32).
            #pragma unroll
            for (int v = 0; v < 8; ++v) {
                p[v] += __shfl_xor(p[v], 1, 32);
                p[v] += __shfl_xor(p[v], 2, 32);
                p[v] += __shfl_xor(p[v], 4, 32);
                p[v] += __shfl_xor(p[v], 8, 32);
            }

            // Park per-row scalars s[row] in this wave's LDS strip.
            #pragma unroll
            for (int v = 0; v < 8; ++v) {
                const int row  = tr0 + v + 8 * hl;
                const float sv = (row < N) ? p[v] : 0.0f;
                if (m == v) s_buf[wave][t * 16 + v + 8 * hl] = sv;
            }
        }

        // Bag pooling: lanes 0..3 each own one bag of 20 scalars.
        if (lane < kBagsPerGroup) {
            const int bagRow0 = r0 + lane * kBagSize;
            if (bagRow0 < N) {
                float ssum = 0.0f;
                #pragma unroll
                for (int j = 0; j < kBagSize; ++j)
                    ssum += s_buf[wave][lane * kBagSize + j];
                const int bag = ids[bagRow0];
                if (bag >= 0 && bag < B)
                    out[bag] = ssum * inv_cnt + bias2;
            }
        }
    }
}

extern "C" void kernel_launch(void* const* d_in, const int* in_sizes, int n_in,
                              void* d_out, int out_size, void* d_ws, size_t ws_size,
                              hipStream_t stream)
{
    const float* x   = (const float*)d_in[0];
    const int*   ids = (const int*)  d_in[1];
    const float* W1  = (const float*)d_in[2];
    const float* b1  = (const float*)d_in[3];
    const float* W2  = (const float*)d_in[4];
    const float* b2  = (const float*)d_in[5];
    float* out = (float*)d_out;

    const int N = in_sizes[0] / kDIn;     // 2,000,000
    const int B = out_size;               // 100,000
    const int ngroups = (N + kRowsPerGroup - 1) / kRowsPerGroup;  // 25,000
    int blocks = (ngroups + kWavesPerBlk - 1) / kWavesPerBlk;     // 3,125
    if (blocks < 1) blocks = 1;

    bag_mil_wmma_kernel<<<blocks, kBlockSize, 0, stream>>>(
        x, ids, W1, b1, W2, b2, out, N, B, ngroups);
}